// EFR_23021024706959
// MI455X (gfx1250) — compile-verified
//
#include <hip/hip_runtime.h>

// Problem constants (from setup_inputs): x:(8,3,4096) f32, W:(64,) f32, k=20
#define B_    8
#define C_    3
#define N_    4096
#define K_    20
#define COUT_ 64
#define NTILES (N_ / 16)   // 256 tiles of 16 points
#define WAVES  4           // waves per workgroup in kernel 1

typedef __attribute__((ext_vector_type(2))) float v2f;
typedef __attribute__((ext_vector_type(8))) float v8f;

__device__ __forceinline__ bool betterPair(float d, int i, float d2, int i2) {
  // reference top_k ordering: value descending, ties -> lower index first
  return (d > d2) || ((d == d2) && (i < i2));
}

// ---------------------------------------------------------------------------
// Kernel 1: fused pairwise-distance (WMMA f32 16x16x4) + top-20 selection.
// One workgroup handles 16 query points; 4 waves stride over the 256 m-tiles.
// The whole per-batch x slice (48 KB) is staged into LDS once via the CDNA5
// async-to-LDS DMA path, so the inner loop does zero VMEM.
// Writes idx_flat = idx + b*N into the int32 region of d_out.
// ---------------------------------------------------------------------------
__global__ __launch_bounds__(WAVES * 32) void EFR_knn_topk(
    const float* __restrict__ x, int* __restrict__ outIdx) {
  __shared__ __align__(16) float sX[C_ * N_];  // 48 KB: x[b,:,:] slice
  __shared__ float sXXQ[16];                   // query norms
  __shared__ float sTile[WAVES][16 * 16];      // per-wave 16x16 neg-dist tile
  __shared__ float sMD[WAVES * 32][K_];        // partial top-k values
  __shared__ int   sMI[WAVES * 32][K_];        // partial top-k indices

  const int blk  = blockIdx.x;
  const int b    = blk / NTILES;
  const int q0   = (blk % NTILES) * 16;
  const int tid  = threadIdx.x;
  const int wave = tid >> 5;
  const int lane = tid & 31;
  const int col  = lane & 15;
  const bool hi  = lane >= 16;

  const float* __restrict__ xb = x + (size_t)b * C_ * N_;

  // ---- async DMA: global -> LDS, 128 lanes x 24 x b128 = 49152 bytes ----
  {
    const unsigned ldsBase = (unsigned)(size_t)(void*)sX;
#pragma unroll
    for (int o = 0; o < (C_ * N_ * 4) / (WAVES * 32 * 16); ++o) {  // 24
      const unsigned byteOff = (unsigned)(o * (WAVES * 32) + tid) * 16u;
      const unsigned ldsAddr = ldsBase + byteOff;
      const unsigned long long ga =
          (unsigned long long)(size_t)((const char*)xb + byteOff);
      asm volatile("global_load_async_to_lds_b128 %0, %1, off"
                   :: "v"(ldsAddr), "v"(ga) : "memory");
    }
    asm volatile("s_wait_asynccnt 0x0" ::: "memory");
    __syncthreads();  // make staged data visible across all 4 waves
  }

  // ---- A fragment (16x4 f32): queries of this tile, K=3 padded to 4 ----
  // layout: lanes 0-15 hold K=0,1 ; lanes 16-31 hold K=2,3 (pad)
  const int q = q0 + col;
  const float qx0 = sX[q], qx1 = sX[N_ + q], qx2 = sX[2 * N_ + q];
  v2f afrag;
  afrag.x = hi ? qx2 : qx0;
  afrag.y = hi ? 0.0f : qx1;
  if (wave == 0 && lane < 16)
    sXXQ[lane] = qx0 * qx0 + qx1 * qx1 + qx2 * qx2;
  __syncthreads();

  float xxq[8];
#pragma unroll
  for (int v = 0; v < 8; ++v) xxq[v] = sXXQ[v + (hi ? 8 : 0)];

  // ---- per-lane sorted top-20 (register resident, fully unrolled) ----
  float bd[K_];
  int   bi[K_];
#pragma unroll
  for (int j = 0; j < K_; ++j) { bd[j] = -1e30f; bi[j] = 0x7fffffff; }

  // ---- stride over m-tiles: WMMA -> neg-dist -> LDS -> selection ----
  // sTile is strictly per-wave and LDS ops of one wave are in-order, so no
  // workgroup barrier is needed inside this loop -- only a compiler fence.
  for (int t = wave; t < NTILES; t += WAVES) {
    const int m0 = t * 16;
    const int p  = m0 + col;
    const float p0 = sX[p], p1 = sX[N_ + p], p2 = sX[2 * N_ + p];
    v2f bfrag;                                   // B is 4x16 (K x N)
    bfrag.x = hi ? p2 : p0;
    bfrag.y = hi ? 0.0f : p1;

    v8f acc = {};
    acc = __builtin_amdgcn_wmma_f32_16x16x4_f32(
        /*neg_a=*/false, afrag, /*neg_b=*/false, bfrag,
        /*c_mod=*/(short)0, acc, /*reuse_a=*/false, /*reuse_b=*/false);

    const float xxp = p0 * p0 + p1 * p1 + p2 * p2;
#pragma unroll
    for (int v = 0; v < 8; ++v) {
      const int row = v + (hi ? 8 : 0);
      sTile[wave][row * 16 + col] = 2.0f * acc[v] - xxq[v] - xxp;
    }
    asm volatile("" ::: "memory");  // keep store->load order; HW is in-order

    // lane owns (row = lane&15, half = lane>=16 ? cols 8..15 : cols 0..7)
    const int row = col;
    const int cb  = hi ? 8 : 0;
#pragma unroll
    for (int i = 0; i < 8; ++i) {
      const float d = sTile[wave][row * 16 + cb + i];
      const int   m = m0 + cb + i;
      if (d > bd[K_ - 1]) {                      // sorted-insert (unrolled)
        float cd = d; int ci = m;
#pragma unroll
        for (int j = 0; j < K_; ++j) {
          const bool sw = cd > bd[j];            // strict: m ascending keeps
          const float td = sw ? bd[j] : cd;      //   tie order = index order
          const int   ti = sw ? bi[j] : ci;
          bd[j] = sw ? cd : bd[j];
          bi[j] = sw ? ci : bi[j];
          cd = td; ci = ti;
        }
      }
    }
    asm volatile("" ::: "memory");  // reads precede next iter's overwrites
  }

  // ---- merge: 8 disjoint partial lists per row -> final top-20 ----
  {
    const int list = tid;
#pragma unroll
    for (int j = 0; j < K_; ++j) { sMD[list][j] = bd[j]; sMI[list][j] = bi[j]; }
  }
  __syncthreads();

  if (tid < 16) {
    const int row = tid;
    float fd[K_];
    int   fi[K_];
#pragma unroll
    for (int j = 0; j < K_; ++j) { fd[j] = -1e30f; fi[j] = 0x7fffffff; }

    for (int w = 0; w < WAVES; ++w) {
      for (int h = 0; h < 2; ++h) {
        const int l = w * 32 + h * 16 + row;
        for (int j = 0; j < K_; ++j) {
          const float d = sMD[l][j];
          const int   m = sMI[l][j];
          if (betterPair(d, m, fd[K_ - 1], fi[K_ - 1])) {
            float cd = d; int ci = m;
#pragma unroll
            for (int u = 0; u < K_; ++u) {
              const bool sw = betterPair(cd, ci, fd[u], fi[u]);
              const float td = sw ? fd[u] : cd;
              const int   ti = sw ? fi[u] : ci;
              fd[u] = sw ? cd : fd[u];
              fi[u] = sw ? ci : fi[u];
              cd = td; ci = ti;
            }
          }
        }
      }
    }

    const int n    = q0 + row;
    const int base = (b * N_ + n) * K_;
#pragma unroll
    for (int j = 0; j < K_; ++j) outIdx[base + j] = fi[j] + b * N_;
  }
}

// ---------------------------------------------------------------------------
// Kernel 2: gather 3 channels per neighbor, mean, fan out x W (64).
// One thread per (b,n,j); per-o stores are coalesced across consecutive j.
// Pure HBM-write-bandwidth bound (~168 MB at 23.3 TB/s).
// ---------------------------------------------------------------------------
__global__ __launch_bounds__(256) void EFR_fanout(
    const float* __restrict__ x, const float* __restrict__ W,
    const int* __restrict__ idxFlat, float* __restrict__ out) {
  __shared__ float sW[COUT_];
  if (threadIdx.x < COUT_) sW[threadIdx.x] = W[threadIdx.x];
  __syncthreads();

  const int t = blockIdx.x * 256 + threadIdx.x;
  if (t >= B_ * N_ * K_) return;

  const int j = t % K_;
  const int n = (t / K_) % N_;
  const int b = t / (K_ * N_);

  const int g  = idxFlat[t];        // idx + b*N
  const int id = g - b * N_;
  const float* __restrict__ xb = x + (size_t)b * C_ * N_;
  const float fm =
      (xb[id] + xb[N_ + id] + xb[2 * N_ + id]) * (1.0f / 3.0f);

  // out[((b*64 + o)*N + n)*K + j]
  const size_t obase = ((size_t)b * COUT_ * N_ + (size_t)n) * K_ + (size_t)j;
#pragma unroll 8
  for (int o = 0; o < COUT_; ++o)
    out[obase + (size_t)o * (N_ * K_)] = sW[o] * fm;
}

extern "C" void kernel_launch(void* const* d_in, const int* in_sizes, int n_in,
                              void* d_out, int out_size, void* d_ws, size_t ws_size,
                              hipStream_t stream) {
  const float* x = (const float*)d_in[0];
  const float* W = (const float*)d_in[1];
  // d_in[2] is k (==20), baked into K_

  float* outF = (float*)d_out;
  const size_t FLOATS = (size_t)B_ * COUT_ * N_ * K_;   // 41,943,040
  int* outIdx = (int*)d_out + FLOATS;                   // int32 region

  EFR_knn_topk<<<B_ * NTILES, WAVES * 32, 0, stream>>>(x, outIdx);
  EFR_fanout<<<(B_ * N_ * K_ + 255) / 256, 256, 0, stream>>>(x, W, outIdx, outF);
}